// SimpleGNN_51960514347376
// MI455X (gfx1250) — compile-verified
//
#include <hip/hip_runtime.h>

#define NNODES 100000
#define NEDGES 1600000
#define IN_C 64
#define HID_C 64
#define OUT_C 32

typedef __attribute__((ext_vector_type(2))) float v2f;
typedef __attribute__((ext_vector_type(8))) float v8f;

// ---------------------------------------------------------------------------
// Degree / normalization kernels (shared across both layers)
// ---------------------------------------------------------------------------
__global__ void deg_init_kernel(float* __restrict__ deg, int n) {
    int i = blockIdx.x * blockDim.x + threadIdx.x;
    if (i < n) deg[i] = 1.0f;  // self-loop contribution
}

__global__ void deg_count_kernel(const long long* __restrict__ dst,
                                 float* __restrict__ deg, int e) {
    int i = blockIdx.x * blockDim.x + threadIdx.x;
    if (i < e) unsafeAtomicAdd(&deg[(int)dst[i]], 1.0f);
}

__global__ void dinv_kernel(float* __restrict__ deg, int n) {
    int i = blockIdx.x * blockDim.x + threadIdx.x;
    if (i < n) deg[i] = rsqrtf(deg[i]);  // deg >= 1, no inf possible
}

// ---------------------------------------------------------------------------
// f32 GEMM via V_WMMA_F32_16X16X4_F32 (exact f32 path, wave32).
// One wave computes one 16x16 tile of H = A[MxK] @ W[KxN] (row-major).
// blockDim = (32, N/16): wave y owns column tile n0 = 16*y.
// A 16x4 f32 layout: lane<16 -> {K=k0,k0+1}; lane>=16 -> {K=k0+2,k0+3}, M=lane&15.
// B 4x16 f32 layout: mirrored (N=lane&15, K split across half-waves).
// C/D 16x16 f32: VGPR r -> row m0 + 8*(lane>>4) + r, col n0 + (lane&15).
// ---------------------------------------------------------------------------
template <int APPLY_RELU, int K, int NC>
__global__ void gemm_wmma_f32_kernel(const float* __restrict__ A,
                                     const float* __restrict__ W,
                                     float* __restrict__ H, int M) {
    const int lane = threadIdx.x;      // 0..31
    const int half = lane >> 4;        // 0 or 1
    const int l    = lane & 15;
    const int m0   = blockIdx.x * 16;
    const int n0   = threadIdx.y * 16;

    v8f acc = {};
#pragma unroll
    for (int k0 = 0; k0 < K; k0 += 4) {
        const int ka = k0 + half * 2;
        // A fragment: row m0+l, cols ka, ka+1
        const float* ap = A + (size_t)(m0 + l) * K + ka;
        float a0 = ap[0], a1 = ap[1];
        if (APPLY_RELU) { a0 = fmaxf(a0, 0.0f); a1 = fmaxf(a1, 0.0f); }
        v2f av = {a0, a1};
        // B fragment: rows ka, ka+1, col n0+l
        const float* bp = W + (size_t)ka * NC + n0 + l;
        v2f bv = {bp[0], bp[NC]};
        acc = __builtin_amdgcn_wmma_f32_16x16x4_f32(
            /*neg_a=*/false, av, /*neg_b=*/false, bv,
            /*c_mod=*/(short)0, acc, /*reuse_a=*/false, /*reuse_b=*/false);
    }

    float* op = H + (size_t)(m0 + half * 8) * NC + n0 + l;
#pragma unroll
    for (int r = 0; r < 8; ++r) op[(size_t)r * NC] = acc[r];
}

// ---------------------------------------------------------------------------
// out[i,c] = bias[c] + h[i,c] * dinv[i]^2   (self-loop term; also clears the
// poisoned output buffer). One thread per 4 channels, float4 traffic.
// ---------------------------------------------------------------------------
template <int C>
__global__ void init_out_kernel(const float* __restrict__ h,
                                const float* __restrict__ dinv,
                                const float* __restrict__ bias,
                                float* __restrict__ out, int n) {
    const int tid = blockIdx.x * blockDim.x + threadIdx.x;
    const int i = tid / (C / 4);
    const int q = (tid % (C / 4)) * 4;
    if (i >= n) return;
    const float di = dinv[i];
    const float s  = di * di;
    const float4 hv = *(const float4*)(h + (size_t)i * C + q);
    float4 ov;
    ov.x = bias[q + 0] + hv.x * s;
    ov.y = bias[q + 1] + hv.y * s;
    ov.z = bias[q + 2] + hv.z * s;
    ov.w = bias[q + 3] + hv.w * s;
    *(float4*)(out + (size_t)i * C + q) = ov;
}

// ---------------------------------------------------------------------------
// Edge scatter: out[dst] += h[src] * dinv[src]*dinv[dst].
// One thread per (edge, 4-channel quad): coalesced 128-bit gather (L2-hot),
// 4x hardware global_atomic_add_f32.
// ---------------------------------------------------------------------------
template <int C>
__global__ void scatter_kernel(const long long* __restrict__ src,
                               const long long* __restrict__ dst,
                               const float* __restrict__ h,
                               const float* __restrict__ dinv,
                               float* __restrict__ out, int e) {
    const int tid = blockIdx.x * blockDim.x + threadIdx.x;
    const int edge = tid / (C / 4);
    const int q    = (tid % (C / 4)) * 4;
    if (edge >= e) return;
    const int s = (int)src[edge];
    const int d = (int)dst[edge];
    const float nrm = dinv[s] * dinv[d];
    const float4 hv = *(const float4*)(h + (size_t)s * C + q);
    float* o = out + (size_t)d * C + q;
    unsafeAtomicAdd(o + 0, hv.x * nrm);
    unsafeAtomicAdd(o + 1, hv.y * nrm);
    unsafeAtomicAdd(o + 2, hv.z * nrm);
    unsafeAtomicAdd(o + 3, hv.w * nrm);
}

// ---------------------------------------------------------------------------
extern "C" void kernel_launch(void* const* d_in, const int* in_sizes, int n_in,
                              void* d_out, int out_size, void* d_ws, size_t ws_size,
                              hipStream_t stream) {
    const float*     x   = (const float*)d_in[0];
    const long long* ei  = (const long long*)d_in[1];  // int64 [2, E]
    const float*     W1  = (const float*)d_in[2];
    const float*     b1  = (const float*)d_in[3];
    const float*     W2  = (const float*)d_in[4];
    const float*     b2  = (const float*)d_in[5];
    float*           out = (float*)d_out;              // [N, 32]

    const long long* src = ei;            // row 0
    const long long* dst = ei + NEDGES;   // row 1

    // Workspace layout (floats): dinv[N] | h1[N*64] | out1[N*64] | h2[N*32]
    float* dinv = (float*)d_ws;
    float* h1   = dinv + NNODES;
    float* out1 = h1 + (size_t)NNODES * HID_C;
    float* h2   = out1 + (size_t)NNODES * HID_C;

    const int B = 256;
    const int Mtiles = NNODES / 16;  // 6250 (N divisible by 16)

    // --- normalization (shared by both layers) ---
    deg_init_kernel<<<(NNODES + B - 1) / B, B, 0, stream>>>(dinv, NNODES);
    deg_count_kernel<<<(NEDGES + B - 1) / B, B, 0, stream>>>(dst, dinv, NEDGES);
    dinv_kernel<<<(NNODES + B - 1) / B, B, 0, stream>>>(dinv, NNODES);

    // --- layer 1: h1 = x @ W1 ; out1 = b1 + self + scatter ---
    gemm_wmma_f32_kernel<0, IN_C, HID_C>
        <<<Mtiles, dim3(32, HID_C / 16), 0, stream>>>(x, W1, h1, NNODES);
    {
        int total = NNODES * (HID_C / 4);
        init_out_kernel<HID_C><<<(total + B - 1) / B, B, 0, stream>>>(h1, dinv, b1, out1, NNODES);
    }
    {
        int total = NEDGES * (HID_C / 4);
        scatter_kernel<HID_C><<<(total + B - 1) / B, B, 0, stream>>>(src, dst, h1, dinv, out1, NEDGES);
    }

    // --- layer 2: h2 = relu(out1) @ W2 ; out = b2 + self + scatter ---
    gemm_wmma_f32_kernel<1, HID_C, OUT_C>
        <<<Mtiles, dim3(32, OUT_C / 16), 0, stream>>>(out1, W2, h2, NNODES);
    {
        int total = NNODES * (OUT_C / 4);
        init_out_kernel<OUT_C><<<(total + B - 1) / B, B, 0, stream>>>(h2, dinv, b2, out, NNODES);
    }
    {
        int total = NEDGES * (OUT_C / 4);
        scatter_kernel<OUT_C><<<(total + B - 1) / B, B, 0, stream>>>(src, dst, h2, dinv, out, NEDGES);
    }
}